// PCasso_23562190586490
// MI455X (gfx1250) — compile-verified
//
#include <hip/hip_runtime.h>
#include <hip/hip_bf16.h>
#include <math.h>

// CDNA5 / gfx1250, wave32. Matrix op: V_WMMA_F32_16X16X4_F32 (full fp32).
// Global->LDS staging uses GLOBAL_LOAD_ASYNC_TO_LDS_B128 (ASYNCcnt) with
// double-buffered LDS tiles; the NN-transpose and scaled-A paths stage
// through VGPRs.
typedef float v2f __attribute__((ext_vector_type(2)));
typedef float v8f __attribute__((ext_vector_type(8)));

#define GAMMA 0.5f
#define BSZ   8192
#define BM    64
#define BN    64
#define BK    32
#define LDP   36   // padded LDS row stride (floats); keeps 16B slots aligned

__device__ __forceinline__ float leaky(float v) { return v > 0.f ? v : 0.2f * v; }

// ---- CDNA5 async global->LDS copy (16B per lane), tracked by ASYNCcnt ----
__device__ __forceinline__ void async_cp16(unsigned lds_addr, const float* gptr) {
    asm volatile("global_load_async_to_lds_b128 %0, %1, off"
                 :: "v"(lds_addr), "v"(gptr) : "memory");
}
__device__ __forceinline__ void wait_async0() {
    asm volatile("s_wait_asynccnt 0x0" ::: "memory");
}
__device__ __forceinline__ unsigned lds_addr_of(const float* p) {
    return (unsigned)(unsigned long long)p;   // generic->LDS: low 32 bits
}

struct GP {
    const float* A;       // [8192, K]
    const float* B;       // weights
    int N, K, transB;     // transB=0: B[k*N+n]; transB=1: B[n*K+k]
    const float* ascale;  // len K, applied to A on load (or null)
    const float* bias;    // len N (or null)
    const float* auxv;    // len N row vector (or null)
    const float* aux0;    // [8192,N] (or null)
    const float* aux1;    // [8192,N] (or null)
    float* out0;
    float* out1;
    float* partial;       // per-block partial sums (EPI>=4)
    float c0;             // 1/(B*d) mean-gradient scale
};

// ---------------------------------------------------------------------------
// Tiny kernel: broadcast feed-forward rows (mu1/x2/x3 are batch-independent).
// ---------------------------------------------------------------------------
__global__ __launch_bounds__(256) void init_rows(
    const float* __restrict__ W1, const float* __restrict__ b1,
    const float* __restrict__ W2, const float* __restrict__ b2,
    const float* __restrict__ W3, const float* __restrict__ b3,
    const float* __restrict__ W4, const float* __restrict__ b4,
    float* __restrict__ row2, float* __restrict__ row3,
    float* __restrict__ d3row, float* __restrict__ x4row, float* __restrict__ d4row)
{
    __shared__ float sx1[256];
    __shared__ float sr2[512];
    __shared__ float sr3[1024];
    const int t = threadIdx.x;

    sx1[t] = leaky(W1[t] + b1[t]);                       // x1 = leaky(s_in@W1.T + b1)
    __syncthreads();

    for (int o = t; o < 512; o += 256) {                 // x2 row (== initial mu2)
        float s = b2[o];
        const float* w = W2 + (size_t)o * 256;
        for (int k = 0; k < 256; ++k) s += sx1[k] * w[k];
        sr2[o] = leaky(s);
    }
    __syncthreads();

    for (int o = t; o < 1024; o += 256) {                // x3 row (== initial mu3)
        float s = b3[o];
        const float* w = W3 + (size_t)o * 512;
        for (int k = 0; k < 512; ++k) s += sr2[k] * w[k];
        sr3[o]   = leaky(s);
        row3[o]  = sr3[o];
        d3row[o] = (s > 0.f) ? 1.f : 0.2f;
    }
    __syncthreads();

    for (int o = t; o < 512; o += 256) row2[o] = sr2[o];
    for (int o = t; o < 784; o += 256) {                 // x4 row of iteration 1
        float s = b4[o];
        const float* w = W4 + (size_t)o * 1024;
        for (int k = 0; k < 1024; ++k) s += sr3[k] * w[k];
        float x4 = tanhf(s);
        x4row[o] = x4;
        d4row[o] = 1.f - x4 * x4;
    }
}

// ---------------------------------------------------------------------------
// Iteration-1 elementwise prep: e4 = (x - x4row)/N4 ; t1 = -e4*d4row ;
// so1 = x - GAMMA*e4*(1-mask)
// ---------------------------------------------------------------------------
__global__ void pre_iter1(const float* __restrict__ x, const float* __restrict__ mask,
                          const float* __restrict__ x4row, const float* __restrict__ d4row,
                          float* __restrict__ t1, float* __restrict__ so1,
                          float inv4, long n)
{
    long i = (long)blockIdx.x * blockDim.x + threadIdx.x;
    if (i >= n) return;
    int col = (int)(i % 784);
    float xv = x[i];
    float e4 = (xv - x4row[col]) * inv4;
    t1[i]  = -e4 * d4row[col];
    so1[i] = xv - GAMMA * e4 * (1.f - mask[i]);
}

// ---------------------------------------------------------------------------
// WMMA f32 GEMM: C[8192,N] = A[8192,K] x B, double-buffered LDS, fused
// epilogues (see EPI table in round-1 comments).
// ---------------------------------------------------------------------------
template <int EPI, bool AASYNC, bool BASYNC>
__global__ __launch_bounds__(128) void gemm_wmma(GP p)
{
    __shared__ float As[2][BM * LDP];
    __shared__ float Bs[2][BN * LDP];
    __shared__ float red[128];

    const int tid  = threadIdx.x;
    const int wave = tid >> 5;
    const int lane = tid & 31;
    const int half = lane >> 4;   // ISA lane split: lanes 0-15 / 16-31
    const int l16  = lane & 15;
    const int m0 = blockIdx.y * BM;
    const int n0 = blockIdx.x * BN;
    const int N = p.N, K = p.K;

    v8f acc[4];
#pragma unroll
    for (int i = 0; i < 4; ++i) acc[i] = (v8f){0.f,0.f,0.f,0.f,0.f,0.f,0.f,0.f};

    // ---- tile loaders (into buffer `bf`) ----
    auto loadA = [&](int k0, int bf) {
#pragma unroll
        for (int i = 0; i < 4; ++i) {
            int idx = tid + i * 128;          // 512 16B slots
            int row = idx >> 3;
            int kk  = (idx & 7) << 2;
            int gk  = k0 + kk;
            float* dst = &As[bf][row * LDP + kk];
            const float* ap = p.A + (size_t)(m0 + row) * K + gk;
            if constexpr (AASYNC) {
                if (gk + 4 <= K) async_cp16(lds_addr_of(dst), ap);   // K%4==0: all-or-none
                else             *(float4*)dst = make_float4(0.f, 0.f, 0.f, 0.f);
            } else {
                float4 v = make_float4(0.f, 0.f, 0.f, 0.f);
                if (gk + 4 <= K) {
                    v = *(const float4*)ap;
                    v.x *= p.ascale[gk + 0]; v.y *= p.ascale[gk + 1];
                    v.z *= p.ascale[gk + 2]; v.w *= p.ascale[gk + 3];
                }
                *(float4*)dst = v;
            }
        }
    };
    auto loadB = [&](int k0, int bf) {
        if constexpr (BASYNC) {               // B[n*K+k], contiguous along k
#pragma unroll
            for (int i = 0; i < 4; ++i) {
                int idx = tid + i * 128;
                int row = idx >> 3;           // n within tile
                int kk  = (idx & 7) << 2;
                int gn = n0 + row, gk = k0 + kk;
                float* dst = &Bs[bf][row * LDP + kk];
                if (gn < N && gk + 4 <= K)
                    async_cp16(lds_addr_of(dst), p.B + (size_t)gn * K + gk);
                else
                    *(float4*)dst = make_float4(0.f, 0.f, 0.f, 0.f);
            }
        } else if (p.transB) {                // staged NT (unused today, kept generic)
#pragma unroll
            for (int i = 0; i < 4; ++i) {
                int idx = tid + i * 128;
                int row = idx >> 3;
                int kk  = (idx & 7) << 2;
                int gn = n0 + row, gk = k0 + kk;
                float4 v = make_float4(0.f, 0.f, 0.f, 0.f);
                if (gn < N && gk + 4 <= K)
                    v = *(const float4*)(p.B + (size_t)gn * K + gk);
                *(float4*)&Bs[bf][row * LDP + kk] = v;
            }
        } else {                              // staged NN: transpose k-major -> Bs[n][k]
#pragma unroll
            for (int i = 0; i < 4; ++i) {
                int idx  = tid + i * 128;
                int krow = idx >> 4;          // k within tile (0..31)
                int nn   = (idx & 15) << 2;   // n within tile
                int gk = k0 + krow;
                float v0 = 0.f, v1 = 0.f, v2 = 0.f, v3 = 0.f;
                if (gk < K) {
                    const float* bp = p.B + (size_t)gk * N + n0 + nn;
                    if (n0 + nn + 0 < N) v0 = bp[0];
                    if (n0 + nn + 1 < N) v1 = bp[1];
                    if (n0 + nn + 2 < N) v2 = bp[2];
                    if (n0 + nn + 3 < N) v3 = bp[3];
                }
                Bs[bf][(nn + 0) * LDP + krow] = v0;
                Bs[bf][(nn + 1) * LDP + krow] = v1;
                Bs[bf][(nn + 2) * LDP + krow] = v2;
                Bs[bf][(nn + 3) * LDP + krow] = v3;
            }
        }
    };

    // ---- double-buffered main loop ----
    const int nchunks = (K + BK - 1) / BK;
    int buf = 0;
    loadA(0, 0);
    loadB(0, 0);
    if constexpr (AASYNC || BASYNC) wait_async0();
    __syncthreads();

    const int arow = wave * 16 + l16;
    for (int c = 0; c < nchunks; ++c) {
        if (c + 1 < nchunks) {                // prefetch next tile into other buffer
            loadA((c + 1) * BK, buf ^ 1);
            loadB((c + 1) * BK, buf ^ 1);
        }
        // 8 WMMA K-steps of 4; A 16x4 / B 4x16 lane layout per ISA §7.12.2
#pragma unroll
        for (int kk = 0; kk < BK; kk += 4) {
            v2f a = *(const v2f*)&As[buf][arow * LDP + kk + half * 2];
#pragma unroll
            for (int nt = 0; nt < 4; ++nt) {
                v2f b = *(const v2f*)&Bs[buf][(nt * 16 + l16) * LDP + kk + half * 2];
                acc[nt] = __builtin_amdgcn_wmma_f32_16x16x4_f32(
                    false, a, false, b, (short)0, acc[nt], false, false);
            }
        }
        if constexpr (AASYNC || BASYNC) wait_async0();  // my async writes landed
        __syncthreads();                                // tile visible to all waves
        buf ^= 1;
    }

    // ---- fused epilogue (C/D layout: row = v + half*8, col = l16) ----
    float lsum = 0.f;
#pragma unroll
    for (int nt = 0; nt < 4; ++nt) {
#pragma unroll
        for (int v = 0; v < 8; ++v) {
            int gm = m0 + wave * 16 + v + half * 8;
            int gn = n0 + nt * 16 + l16;
            if (gn >= N) continue;
            size_t o = (size_t)gm * N + gn;
            float a = acc[nt][v];
            if constexpr (EPI == 1) {
                p.out0[o] = p.auxv[gn] - GAMMA * a;      // mu3 after iter1
                p.out1[o] = -GAMMA * a * p.c0;           // e3 for iter2
            } else if constexpr (EPI == 2) {
                float x4 = tanhf(a + p.bias[gn]);
                float s1 = p.aux0[o];
                float e4 = (s1 - x4) * p.c0;
                p.out0[o] = -e4 * (1.f - x4 * x4);       // t4
                p.out1[o] = s1 - GAMMA * e4 * (1.f - p.aux1[o]);  // final s_out
            } else if constexpr (EPI == 3) {
                p.out0[o] = p.aux1[o] - GAMMA * (p.aux0[o] + a);  // mu3 final
            } else if constexpr (EPI == 4) {
                float d = GAMMA * a;
                p.out0[o] = p.auxv[gn] + d;              // mu2 final
                lsum += d * d;                           // term2
            } else if constexpr (EPI == 5) {
                float d = leaky(a + p.bias[gn]) - p.aux0[o];
                lsum += d * d;                           // term3
            } else if constexpr (EPI == 6) {
                float d = tanhf(a + p.bias[gn]) - p.aux0[o];
                lsum += d * d;                           // term4
            }
        }
    }
    if constexpr (EPI >= 4) {
        red[tid] = lsum;
        __syncthreads();
        for (int s = 64; s > 0; s >>= 1) {
            if (tid < s) red[tid] += red[tid + s];
            __syncthreads();
        }
        if (tid == 0) p.partial[blockIdx.y * gridDim.x + blockIdx.x] = red[0];
    }
}

// ---------------------------------------------------------------------------
// Deterministic final reduction: E = 0.5*(0 + t2/N2 + t3/N3 + t4/N4)
// ---------------------------------------------------------------------------
__global__ __launch_bounds__(256) void final_energy(
    const float* __restrict__ p2, const float* __restrict__ p3,
    const float* __restrict__ p4, int n2, int n3, int n4, float* __restrict__ out)
{
    __shared__ float r[768];
    const int t = threadIdx.x;
    float s2 = 0.f, s3 = 0.f, s4 = 0.f;
    for (int i = t; i < n2; i += 256) s2 += p2[i];
    for (int i = t; i < n3; i += 256) s3 += p3[i];
    for (int i = t; i < n4; i += 256) s4 += p4[i];
    r[t] = s2; r[256 + t] = s3; r[512 + t] = s4;
    __syncthreads();
    for (int s = 128; s > 0; s >>= 1) {
        if (t < s) { r[t] += r[t + s]; r[256 + t] += r[256 + t + s]; r[512 + t] += r[512 + t + s]; }
        __syncthreads();
    }
    if (t == 0)
        out[0] = 0.5f * (r[0]   / (8192.f * 512.f)
                       + r[256] / (8192.f * 1024.f)
                       + r[512] / (8192.f * 784.f));
}

// ---------------------------------------------------------------------------
extern "C" void kernel_launch(void* const* d_in, const int* in_sizes, int n_in,
                              void* d_out, int out_size, void* d_ws, size_t ws_size,
                              hipStream_t stream)
{
    (void)in_sizes; (void)n_in; (void)out_size; (void)ws_size;
    const float* x    = (const float*)d_in[0];
    const float* mask = (const float*)d_in[1];
    const float* W1 = (const float*)d_in[2]; const float* b1 = (const float*)d_in[3];
    const float* W2 = (const float*)d_in[4]; const float* b2 = (const float*)d_in[5];
    const float* W3 = (const float*)d_in[6]; const float* b3 = (const float*)d_in[7];
    const float* W4 = (const float*)d_in[8]; const float* b4 = (const float*)d_in[9];
    float* out = (float*)d_out;

    float* ws    = (float*)d_ws;
    float* row2  = ws;          // 512
    float* row3  = ws + 512;    // 1024
    float* d3row = ws + 1536;   // 1024
    float* x4row = ws + 2560;   // 784
    float* d4row = ws + 3344;   // 784
    float* p2    = ws + 4128;   // 1024 partials
    float* p3    = ws + 5152;   // 2048 partials
    float* p4    = ws + 7200;   // 1664 partials
    const size_t B784  = (size_t)BSZ * 784;
    const size_t B1024 = (size_t)BSZ * 1024;
    const size_t B512  = (size_t)BSZ * 512;
    float* T    = ws + 16384;   // [B,784]  t1, later reused as t4
    float* SO1  = T    + B784;  // [B,784]
    float* MU3  = SO1  + B784;  // [B,1024]
    float* E3   = MU3  + B1024; // [B,1024]
    float* MU2F = E3   + B1024; // [B,512]
    float* MU3F = MU2F + B512;  // [B,1024]
    float* soF  = out + 1;      // final s_out written straight into d_out

    init_rows<<<1, 256, 0, stream>>>(W1, b1, W2, b2, W3, b3, W4, b4,
                                     row2, row3, d3row, x4row, d4row);

    const long n = (long)BSZ * 784;
    pre_iter1<<<(unsigned)((n + 255) / 256), 256, 0, stream>>>(
        x, mask, x4row, d4row, T, SO1, 1.f / (8192.f * 784.f), n);

    dim3 blk(128);
    auto gridFor = [](int N) { return dim3((unsigned)((N + BN - 1) / BN), BSZ / BM); };

    {   // G1: mu3/e3 from g3 = t1 @ W4   [B,784]x[784,1024] (NN)
        GP g{}; g.A = T; g.B = W4; g.N = 1024; g.K = 784; g.transB = 0;
        g.auxv = row3; g.out0 = MU3; g.out1 = E3; g.c0 = 1.f / (8192.f * 1024.f);
        gemm_wmma<1, true, false><<<gridFor(1024), blk, 0, stream>>>(g);
    }
    {   // G2: iter2 forward z4 = mu3 @ W4.T + b4   [B,1024]x[1024,784] (NT)
        GP g{}; g.A = MU3; g.B = W4; g.N = 784; g.K = 1024; g.transB = 1;
        g.bias = b4; g.aux0 = SO1; g.aux1 = mask; g.out0 = T; g.out1 = soF;
        g.c0 = 1.f / (8192.f * 784.f);
        gemm_wmma<2, true, true><<<gridFor(784), blk, 0, stream>>>(g);
    }
    {   // G3: mu3f = mu3 - g*(e3 + t4 @ W4)   (NN)
        GP g{}; g.A = T; g.B = W4; g.N = 1024; g.K = 784; g.transB = 0;
        g.aux0 = E3; g.aux1 = MU3; g.out0 = MU3F;
        gemm_wmma<3, true, false><<<gridFor(1024), blk, 0, stream>>>(g);
    }
    {   // G4: mu2f = row2 + g*((e3*d3row) @ W3), term2   (NN, A scaled on load)
        GP g{}; g.A = E3; g.B = W3; g.N = 512; g.K = 1024; g.transB = 0;
        g.ascale = d3row; g.auxv = row2; g.out0 = MU2F; g.partial = p2;
        gemm_wmma<4, false, false><<<gridFor(512), blk, 0, stream>>>(g);
    }
    {   // G5: term3 from x3 = leaky(mu2f @ W3.T + b3)   (NT)
        GP g{}; g.A = MU2F; g.B = W3; g.N = 1024; g.K = 512; g.transB = 1;
        g.bias = b3; g.aux0 = MU3F; g.partial = p3;
        gemm_wmma<5, true, true><<<gridFor(1024), blk, 0, stream>>>(g);
    }
    {   // G6: term4 from x4 = tanh(mu3f @ W4.T + b4)   (NT)
        GP g{}; g.A = MU3F; g.B = W4; g.N = 784; g.K = 1024; g.transB = 1;
        g.bias = b4; g.aux0 = soF; g.partial = p4;
        gemm_wmma<6, true, true><<<gridFor(784), blk, 0, stream>>>(g);
    }

    final_energy<<<1, 256, 0, stream>>>(p2, p3, p4, 8 * 128, 16 * 128, 13 * 128, out);
}